// AutoencoderBlock_75522704933234
// MI455X (gfx1250) — compile-verified
//
#include <hip/hip_runtime.h>
#include <hip/hip_bf16.h>
#include <math.h>

// ---------------- problem constants (from reference) ----------------
#define BB    4
#define TT    2048
#define DD    256
#define SRK   128
#define RANKR 8
#define LFFT  4096      // next pow2 >= 2T-1
#define NALPHA 8
#define EPSF  1e-5f

typedef float v2f __attribute__((ext_vector_type(2)));
typedef float v8f __attribute__((ext_vector_type(8)));

// ---------------- helpers ----------------
__device__ inline float blockReduce256(float v, float* red) {
  int t = threadIdx.x;
  red[t] = v; __syncthreads();
  for (int s = 128; s > 0; s >>= 1) { if (t < s) red[t] += red[t + s]; __syncthreads(); }
  float r = red[0]; __syncthreads();
  return r;
}

// In-LDS Stockham radix-2 FFT, N = 4096 (12 stages, even -> result ends in x).
// sgn = -1 forward, +1 inverse (no 1/N scaling; folded in by caller).
__device__ inline void fft_lds(float2* x, float2* y, int tid, int nthr, float sgn) {
  float2* src = x; float2* dst = y;
  int sShift = 0;
  for (int nn = LFFT; nn > 1; nn >>= 1) {
    int m = nn >> 1;
    int s = 1 << sShift;
    float th0 = sgn * 6.28318530717958647692f / (float)nn;
    for (int idx = tid; idx < (LFFT >> 1); idx += nthr) {
      int p = idx >> sShift;
      int q = idx & (s - 1);
      float cw, sw; __sincosf(th0 * (float)p, &sw, &cw);
      float2 a = src[q + (p << sShift)];
      float2 b = src[q + ((p + m) << sShift)];
      float2 e; e.x = a.x + b.x; e.y = a.y + b.y;
      float rx = a.x - b.x, ry = a.y - b.y;
      float2 o; o.x = rx * cw - ry * sw; o.y = rx * sw + ry * cw;
      dst[q + ((2 * p) << sShift)]     = e;
      dst[q + ((2 * p + 1) << sShift)] = o;
    }
    __syncthreads();
    float2* t2 = src; src = dst; dst = t2;
    sShift++;
  }
}

// ---------------- FrFT chirp / H-hat tables (double, mirrors numpy) ----------------
// ai in [0,16): ai<8 -> +alpha_i, ai>=8 -> -alpha_{ai-8}
__global__ void fill_tables_kernel(float2* preT, float2* Hh, float2* prefA) {
  int ai = blockIdx.x;
  int k  = ai & 7;
  double alpha = (0.15 + (double)k * ((2.99 - 0.15) / 7.0)) * (ai < 8 ? 1.0 : -1.0);
  const double PI = 3.14159265358979323846;
  double a = fmod(alpha + PI, 2.0 * PI);
  if (a < 0.0) a += 2.0 * PI;
  a -= PI;
  double sa = sin(a);
  double absa = fabs(sa); if (absa < 1e-7) absa = 1e-7;
  double s_ = (sa >= 0.0 ? 1.0 : -1.0) / absa;        // copysign(1/max(1e-7,|sa|), sa)
  double den = sa; if (den < 1e-7) den = 1e-7;        // python: max(1e-7, sa)  (quirk kept)
  double c_ = cos(a) / den;
  double dtd = 2.0 / (double)(TT - 1);
  for (int j = threadIdx.x; j < TT; j += blockDim.x) {
    double t = -1.0 + (double)j * dtd;
    double sn, cn; sincos(PI * (c_ + s_) * t * t, &sn, &cn);
    float2 v; v.x = (float)cn; v.y = (float)sn;
    preT[(size_t)ai * TT + j] = v;
  }
  for (int i = threadIdx.x; i < LFFT; i += blockDim.x) {
    float2 v; v.x = 0.f; v.y = 0.f;
    long m = 0; bool has = false;
    if (i <= TT - 1) { m = i; has = true; }
    else if (i >= LFFT - (TT - 1)) { m = (long)i - LFFT; has = true; }
    if (has) {
      double md = (double)m * dtd;
      double sn, cn; sincos(-PI * s_ * md * md, &sn, &cn);
      v.x = (float)cn; v.y = (float)sn;
    }
    Hh[(size_t)ai * LFFT + i] = v;
  }
  if (threadIdx.x == 0) {
    // pref = sqrt(1 - i*c_), principal branch (re part of operand = 1 > 0)
    double br = 1.0, bi = -c_;
    double r  = sqrt(br * br + bi * bi);
    double re = sqrt((r + br) * 0.5);
    double im = (re > 0.0) ? bi / (2.0 * re) : 0.0;
    float2 v; v.x = (float)re; v.y = (float)im;
    prefA[ai] = v;
  }
}

// FFT the 16 h_pad tables in place (np.fft.fft, forward sign).
__global__ void fft_table_kernel(float2* Hh) {
  extern __shared__ float2 sm[];
  float2* bx = sm; float2* by = sm + LFFT;
  size_t base = (size_t)blockIdx.x * LFFT;
  for (int i = threadIdx.x; i < LFFT; i += blockDim.x) bx[i] = Hh[base + i];
  __syncthreads();
  fft_lds(bx, by, threadIdx.x, blockDim.x, -1.f);
  for (int i = threadIdx.x; i < LFFT; i += blockDim.x) Hh[base + i] = bx[i];
}

// ---------------- FrFT over one (b, channel) column ----------------
// mode 0: store complex to out_c        mode 1: out_acc += scale * real
// mode 2: out_e = |X|^2  (per-element, deterministic; reduced later)
__global__ void frft_kernel(const float* in_r, const float2* in_c, int input_complex,
                            const float* wvec, int use_w,
                            const float2* pre, const float2* Hh, const float2* prefp,
                            int C, int mode,
                            float2* out_c, float* out_acc, float scale, float* out_e) {
  extern __shared__ float2 sm[];
  float2* bx = sm; float2* by = sm + LFFT;
  int col = blockIdx.x;
  int b = col / C, ch = col - b * C;
  int tid = threadIdx.x, nthr = blockDim.x;
  const float dtf = 2.0f / (float)(TT - 1);
  size_t base = ((size_t)b * TT) * C + ch;
  for (int i = tid; i < LFFT; i += nthr) {
    float2 v; v.x = 0.f; v.y = 0.f;
    if (i < TT) {
      float2 z;
      if (input_complex) z = in_c[base + (size_t)i * C];
      else { z.x = in_r[base + (size_t)i * C]; z.y = 0.f; }
      if (use_w) { float wv = wvec[b * TT + i]; z.x *= wv; z.y *= wv; }
      float2 p = pre[i];
      v.x = z.x * p.x - z.y * p.y;
      v.y = z.x * p.y + z.y * p.x;
    }
    bx[i] = v;
  }
  __syncthreads();
  fft_lds(bx, by, tid, nthr, -1.f);
  for (int i = tid; i < LFFT; i += nthr) {
    float2 a2 = bx[i], h = Hh[i];
    float2 r; r.x = a2.x * h.x - a2.y * h.y; r.y = a2.x * h.y + a2.y * h.x;
    bx[i] = r;
  }
  __syncthreads();
  fft_lds(bx, by, tid, nthr, 1.f);
  float2 pf = *prefp;
  const float sc = dtf / (float)LFFT;        // ifft 1/L scaling folded with dt
  for (int t = tid; t < TT; t += nthr) {
    float2 cv = bx[TT - 1 + t];
    float2 p  = pre[t];
    float2 pp; pp.x = pf.x * p.x - pf.y * p.y; pp.y = pf.x * p.y + pf.y * p.x;
    float2 r;  r.x = (pp.x * cv.x - pp.y * cv.y) * sc;
               r.y = (pp.x * cv.y + pp.y * cv.x) * sc;
    size_t oidx = base + (size_t)t * C;
    if (mode == 0)      out_c[oidx] = r;
    else if (mode == 1) out_acc[oidx] += scale * r.x;
    else                out_e[oidx] = r.x * r.x + r.y * r.y;
  }
}

// mean over channels of |X|^2  ->  E[b*T+t]
__global__ void ereduce_kernel(const float* esc, float* Eout) {
  int idx = blockIdx.x * blockDim.x + threadIdx.x;   // b*T + t
  if (idx >= BB * TT) return;
  const float* p = esc + (size_t)idx * DD;
  float s = 0.f;
  for (int c = 0; c < DD; c++) s += p[c];
  Eout[idx] = s / (float)DD;
}

// w = sqrt(E+1e-6) / (mean_t sqrt(E+1e-6) + 1e-6), per (alpha, b)
__global__ void wnorm_kernel(const float* E, float* W) {
  __shared__ float red[256];
  size_t base = (size_t)blockIdx.x * TT;
  float part = 0.f;
  for (int t = threadIdx.x; t < TT; t += blockDim.x)
    part += sqrtf(E[base + t] + 1e-6f);
  float mean = blockReduce256(part, red) / (float)TT;
  for (int t = threadIdx.x; t < TT; t += blockDim.x)
    W[base + t] = sqrtf(E[base + t] + 1e-6f) / (mean + 1e-6f);
}

// ---------------- fp32 WMMA GEMM: 32x32 C tile per wave (2x2 blocking) ----------
// C[M,N] = A[M,K] * W^T where W is [N,K] row-major (B[k][n] = W[n*ldw + k]).
// mode 0: Cout = gelu(C)                         (exact gelu)
// mode 1: s = C + bias[n]; Cout=s; out2 = addsrc - (n==0) + s
// mode 2: Cout = addsrc + C
__device__ inline void gemm_epilogue(float val, int row, int colj, int N, int mode,
                                     const float* __restrict__ bias,
                                     const float* __restrict__ addsrc,
                                     float* __restrict__ Cout,
                                     float* __restrict__ out2) {
  size_t o = (size_t)row * N + colj;
  if (mode == 0) {
    Cout[o] = 0.5f * val * (1.f + erff(val * 0.70710678118654752f));
  } else if (mode == 1) {
    float sv = val + bias[colj];
    Cout[o] = sv;
    out2[o] = addsrc[o] - (colj == 0 ? 1.f : 0.f) + sv;
  } else {
    Cout[o] = addsrc[o] + val;
  }
}

__global__ void wmma_gemm_kernel(const float* __restrict__ A, const float* __restrict__ W,
                                 float* __restrict__ Cout,
                                 int M, int N, int K, int lda, int ldw,
                                 int mode, const float* __restrict__ bias,
                                 const float* __restrict__ addsrc, float* __restrict__ out2) {
  int wave = blockIdx.x * (blockDim.x >> 5) + (threadIdx.x >> 5);
  int lane = threadIdx.x & 31;
  int tilesN = N >> 5;              // 32-wide C tiles
  int tm = (wave / tilesN) << 5;
  int tn = (wave % tilesN) << 5;
  if (tm >= M) return;
  int half = lane >> 4;             // selects K pair {0,1}/{2,3} in frags, row-half in C
  int l16  = lane & 15;
  v8f c00 = {0.f,0.f,0.f,0.f,0.f,0.f,0.f,0.f};
  v8f c01 = c00, c10 = c00, c11 = c00;
  const float* arow0 = A + (size_t)(tm + l16) * lda;
  const float* arow1 = arow0 + (size_t)16 * lda;
  const float* wrow0 = W + (size_t)(tn + l16) * ldw;
  const float* wrow1 = wrow0 + (size_t)16 * ldw;
  for (int k = 0; k < K; k += 4) {
    int ka = k + (half << 1);
    // prefetch one cacheline ahead on each stream (global_prefetch_b8)
    __builtin_prefetch(arow0 + ka + 32, 0, 1);
    __builtin_prefetch(arow1 + ka + 32, 0, 1);
    v2f a0, a1, b0, b1;
    a0[0] = arow0[ka]; a0[1] = arow0[ka + 1];        // contiguous pair -> b64
    a1[0] = arow1[ka]; a1[1] = arow1[ka + 1];
    b0[0] = wrow0[ka]; b0[1] = wrow0[ka + 1];        // contiguous pair -> b64
    b1[0] = wrow1[ka]; b1[1] = wrow1[ka + 1];
    c00 = __builtin_amdgcn_wmma_f32_16x16x4_f32(false, a0, false, b0, (short)0, c00, false, false);
    c01 = __builtin_amdgcn_wmma_f32_16x16x4_f32(false, a0, false, b1, (short)0, c01, false, false);
    c10 = __builtin_amdgcn_wmma_f32_16x16x4_f32(false, a1, false, b0, (short)0, c10, false, false);
    c11 = __builtin_amdgcn_wmma_f32_16x16x4_f32(false, a1, false, b1, (short)0, c11, false, false);
  }
  for (int v = 0; v < 8; v++) {
    int r0 = tm + v + (half << 3);
    int r1 = r0 + 16;
    int n0 = tn + l16;
    int n1 = n0 + 16;
    gemm_epilogue(c00[v], r0, n0, N, mode, bias, addsrc, Cout, out2);
    gemm_epilogue(c01[v], r0, n1, N, mode, bias, addsrc, Cout, out2);
    gemm_epilogue(c10[v], r1, n0, N, mode, bias, addsrc, Cout, out2);
    gemm_epilogue(c11[v], r1, n1, N, mode, bias, addsrc, Cout, out2);
  }
}

// ---------------- small linear algebra ----------------
__global__ void initv_kernel(float* V) {
  int idx = blockIdx.x * blockDim.x + threadIdx.x;
  if (idx >= BB * DD * RANKR) return;
  int r = idx % RANKR, d = (idx / RANKR) % DD;
  V[idx] = (d == r) ? 1.f : 0.f;
}

__global__ void zero_kernel(float* p, int n) {
  int i = blockIdx.x * blockDim.x + threadIdx.x;
  if (i < n) p[i] = 0.f;
}

// Y[b,t,r] = sum_d xprime[b,t,d] * V[b,d,r]
__global__ void proj_kernel(const float* xp, const float* V, float* Y) {
  int idx = blockIdx.x * blockDim.x + threadIdx.x;  // (b,t)
  if (idx >= BB * TT) return;
  int b = idx / TT;
  float acc[RANKR];
  for (int r = 0; r < RANKR; r++) acc[r] = 0.f;
  const float* xr = xp + (size_t)idx * DD;
  const float* Vb = V + (size_t)b * DD * RANKR;
  for (int d = 0; d < DD; d++) {
    float xv = xr[d];
    const float* vr = Vb + d * RANKR;
    for (int r = 0; r < RANKR; r++) acc[r] += xv * vr[r];
  }
  float* yo = Y + (size_t)idx * RANKR;
  for (int r = 0; r < RANKR; r++) yo[r] = acc[r];
}

// Z[b,d,r] = (1/T) sum_t xp[b,t,d]*Y[b,t,r] + EPS*V[b,d,r]
__global__ void zmat_kernel(const float* xp, const float* Y, const float* V, float* Z) {
  int idx = blockIdx.x * blockDim.x + threadIdx.x;
  if (idx >= BB * DD * RANKR) return;
  int r = idx % RANKR;
  int d = (idx / RANKR) % DD;
  int b = idx / (RANKR * DD);
  float s = 0.f;
  for (int t = 0; t < TT; t++)
    s += xp[((size_t)b * TT + t) * DD + d] * Y[((size_t)b * TT + t) * RANKR + r];
  Z[idx] = s / (float)TT + EPSF * V[idx];
}

// Modified Gram-Schmidt (== jnp QR with positive-diag sign fix); optional anchor sign-align.
__global__ void orthonorm_kernel(const float* Z, float* V, int anchorFlag) {
  __shared__ float red[256];
  __shared__ float bc;
  int b = blockIdx.x, d = threadIdx.x;
  float col[RANKR];
  for (int r = 0; r < RANKR; r++) col[r] = Z[((size_t)b * DD + d) * RANKR + r];
  for (int j = 0; j < RANKR; j++) {
    for (int i = 0; i < j; i++) {
      float dot = blockReduce256(col[i] * col[j], red);
      col[j] -= dot * col[i];
    }
    float nr = blockReduce256(col[j] * col[j], red);
    nr = sqrtf(fmaxf(nr, 1e-20f));
    col[j] /= nr;
    if (anchorFlag) {
      if (d == 0) bc = col[j];
      __syncthreads();
      float sg = (bc + 1e-12f) >= 0.f ? 1.f : -1.f;
      col[j] *= sg;
      __syncthreads();
    }
  }
  for (int r = 0; r < RANKR; r++) V[((size_t)b * DD + d) * RANKR + r] = col[r];
}

// x_hat[b,t,d] = sum_r traces[b,t,r]*V[b,d,r] - s[b,t,d] + (d==0)
__global__ void xhat_kernel(const float* Y, const float* V, const float* sb, float* xh) {
  size_t idx = (size_t)blockIdx.x * blockDim.x + threadIdx.x;
  if (idx >= (size_t)BB * TT * DD) return;
  int d = (int)(idx % DD);
  size_t bt = idx / DD;
  int b = (int)(bt / TT);
  const float* yr = Y + bt * RANKR;
  const float* vr = V + ((size_t)b * DD + d) * RANKR;
  float s = 0.f;
  for (int r = 0; r < RANKR; r++) s += yr[r] * vr[r];
  xh[idx] = s - sb[idx] + (d == 0 ? 1.f : 0.f);
}

// LayerNorm over D
__global__ void ln_kernel(const float* y, const float* g, const float* be, float* out) {
  __shared__ float red[256];
  size_t row = blockIdx.x;
  int d = threadIdx.x;
  float v = y[row * DD + d];
  float mu = blockReduce256(v, red) / (float)DD;
  float dv = v - mu;
  float var = blockReduce256(dv * dv, red) / (float)DD;
  out[row * DD + d] = dv * rsqrtf(var + 1e-5f) * g[d] + be[d];
}

// ---------------- driver ----------------
extern "C" void kernel_launch(void* const* d_in, const int* in_sizes, int n_in,
                              void* d_out, int out_size, void* d_ws, size_t ws_size,
                              hipStream_t stream) {
  (void)in_sizes; (void)n_in; (void)out_size; (void)ws_size;
  const float* x    = (const float*)d_in[0];
  const float* Win  = (const float*)d_in[1];   // [SR, D]
  const float* Wout = (const float*)d_in[2];   // [D, SR]
  const float* bsh  = (const float*)d_in[3];   // [D]
  const float* WO   = (const float*)d_in[4];   // [D, D]
  const float* lng  = (const float*)d_in[5];
  const float* lnb  = (const float*)d_in[6];
  float* out = (float*)d_out;
  float* ws  = (float*)d_ws;

  size_t o = 0;
  float*  Hbuf   = ws + o; o += (size_t)BB * TT * SRK;        // 4 MB
  float*  sbuf   = ws + o; o += (size_t)BB * TT * DD;         // 8 MB
  float*  xprime = ws + o; o += (size_t)BB * TT * DD;         // 8 MB
  float2* preT   = (float2*)(ws + o); o += (size_t)16 * TT * 2;
  float2* Hhat   = (float2*)(ws + o); o += (size_t)16 * LFFT * 2;
  float2* prefA  = (float2*)(ws + o); o += 32;
  float*  Ebuf   = ws + o; o += (size_t)NALPHA * BB * TT;
  float*  wbuf   = ws + o; o += (size_t)NALPHA * BB * TT;
  float*  Ybuf   = ws + o; o += (size_t)BB * TT * RANKR;
  float*  accb   = ws + o; o += (size_t)BB * TT * RANKR;
  float2* tmpC   = (float2*)(ws + o); o += (size_t)BB * TT * RANKR * 2;
  float*  Zbuf   = ws + o; o += (size_t)BB * DD * RANKR;
  float*  Vbuf   = ws + o; o += (size_t)BB * DD * RANKR;
  float*  xhat   = ws + o; o += (size_t)BB * TT * DD;         // 8 MB
  float*  ybuf   = ws + o; o += (size_t)BB * TT * DD;         // 8 MB
  float*  esc    = ybuf;  // |X|^2 scratch aliases ybuf (used strictly earlier)

  const size_t FFT_LDS = (size_t)2 * LFFT * sizeof(float2);   // 64 KB

  // --- FrFT tables (16 signed alphas) ---
  fill_tables_kernel<<<16, 256, 0, stream>>>(preT, Hhat, prefA);
  fft_table_kernel<<<16, 256, FFT_LDS, stream>>>(Hhat);

  // --- low-rank shift: H = gelu(x @ Win^T); s = H @ Wout^T + b; xprime = x - anchor + s ---
  wmma_gemm_kernel<<<(BB*TT/32)*(SRK/32)/8, 256, 0, stream>>>(
      x, Win, Hbuf, BB*TT, SRK, DD, DD, DD, 0, nullptr, nullptr, nullptr);
  wmma_gemm_kernel<<<(BB*TT/32)*(DD/32)/8, 256, 0, stream>>>(
      Hbuf, Wout, sbuf, BB*TT, DD, SRK, SRK, SRK, 1, bsh, x, xprime);

  // --- per-alpha FrFT-domain energy weights ---
  for (int a = 0; a < NALPHA; a++) {
    frft_kernel<<<BB*DD, 256, FFT_LDS, stream>>>(
        xprime, nullptr, 0, nullptr, 0,
        preT + (size_t)a * TT, Hhat + (size_t)a * LFFT, prefA + a,
        DD, 2, nullptr, nullptr, 0.f, esc);
    ereduce_kernel<<<(BB*TT + 255) / 256, 256, 0, stream>>>(esc, Ebuf + (size_t)a * BB * TT);
  }
  wnorm_kernel<<<NALPHA * BB, 256, 0, stream>>>(Ebuf, wbuf);

  // --- subspace iteration ---
  initv_kernel<<<(BB*DD*RANKR + 255) / 256, 256, 0, stream>>>(Vbuf);
  for (int it = 0; it < 2; it++) {
    proj_kernel<<<(BB*TT + 255) / 256, 256, 0, stream>>>(xprime, Vbuf, Ybuf);
    zero_kernel<<<(BB*TT*RANKR + 255) / 256, 256, 0, stream>>>(accb, BB*TT*RANKR);
    for (int a = 0; a < NALPHA; a++) {
      frft_kernel<<<BB*RANKR, 256, FFT_LDS, stream>>>(          // Ya = frft(Yc, +a)
          Ybuf, nullptr, 0, nullptr, 0,
          preT + (size_t)a * TT, Hhat + (size_t)a * LFFT, prefA + a,
          RANKR, 0, tmpC, nullptr, 0.f, nullptr);
      frft_kernel<<<BB*RANKR, 256, FFT_LDS, stream>>>(          // acc += frft(w*Ya, -a)/8
          nullptr, tmpC, 1, wbuf + (size_t)a * BB * TT, 1,
          preT + (size_t)(NALPHA + a) * TT, Hhat + (size_t)(NALPHA + a) * LFFT,
          prefA + NALPHA + a,
          RANKR, 1, nullptr, accb, 1.f / (float)NALPHA, nullptr);
    }
    zmat_kernel<<<(BB*DD*RANKR + 255) / 256, 256, 0, stream>>>(xprime, accb, Vbuf, Zbuf);
    orthonorm_kernel<<<BB, 256, 0, stream>>>(Zbuf, Vbuf, (it == 1) ? 1 : 0);
  }

  // --- reconstruction: traces (== traces_final), x_tilde, x_hat, output ---
  proj_kernel<<<(BB*TT + 255) / 256, 256, 0, stream>>>(xprime, Vbuf, Ybuf);
  xhat_kernel<<<(BB*TT*DD) / 256, 256, 0, stream>>>(Ybuf, Vbuf, sbuf, xhat);
  wmma_gemm_kernel<<<(BB*TT/32)*(DD/32)/8, 256, 0, stream>>>(
      xhat, WO, ybuf, BB*TT, DD, DD, DD, DD, 2, nullptr, x, nullptr);
  ln_kernel<<<BB*TT, 256, 0, stream>>>(ybuf, lng, lnb, out);
}